// MACEConvolutionLayer_88587995447414
// MI455X (gfx1250) — compile-verified
//
#include <hip/hip_runtime.h>
#include <math.h>

typedef __attribute__((ext_vector_type(2))) float v2f;
typedef __attribute__((ext_vector_type(8))) float v8f;

#define NNODES 10000
#define NEDGES 64000
#define MULC   32
#define HIDC   288
#define RHIDC  64
#define SQM    5.656854249492380f   // sqrt(32)

__constant__ int   c_dims[3]  = {1, 3, 5};
__constant__ int   c_off[3]   = {0, 32, 128};
__constant__ int   c_shoff[3] = {0, 1, 4};
__constant__ float c_dinv[3]  = {1.0f, 0.5773502691896258f, 0.4472135954999579f}; // 1/sqrt(d)

// O2_UVU paths: (i,j,k) and CG offsets
__constant__ int c_uvu_i[5]  = {0, 1, 1, 2, 2};
__constant__ int c_uvu_j[5]  = {0, 1, 1, 2, 2};
__constant__ int c_uvu_k[5]  = {0, 0, 2, 0, 2};
__constant__ int c_uvu_cg[5] = {0, 44, 53, 213, 238};

// ---------------------------------------------------------------------------
// CG coefficient table (363 floats, dense per (i,j,k) combo), built on device.
// offsets: (0,0,0)=0 (0,1,1)=1 (0,2,2)=10 (1,0,1)=35 (1,1,0)=44 (1,1,2)=53
//          (1,2,1)=98 (2,0,2)=143 (2,1,1)=168 (2,2,0)=213 (2,2,2)=238
// ---------------------------------------------------------------------------
__global__ void k_init_cg(float* cg) {
    if (threadIdx.x != 0 || blockIdx.x != 0) return;
    const float s2 = sqrtf(2.f), s3 = sqrtf(3.f), s5 = sqrtf(5.f), s6 = sqrtf(6.f);
    float B[5][3][3];
    for (int a = 0; a < 5; a++) for (int i = 0; i < 3; i++) for (int j = 0; j < 3; j++) B[a][i][j] = 0.f;
    B[0][0][1] = B[0][1][0] = 1.f / s2;
    B[1][1][2] = B[1][2][1] = 1.f / s2;
    B[2][0][0] = -1.f / s6; B[2][1][1] = -1.f / s6; B[2][2][2] = 2.f / s6;
    B[3][0][2] = B[3][2][0] = 1.f / s2;
    B[4][0][0] = 1.f / s2;  B[4][1][1] = -1.f / s2;
    for (int q = 0; q < 363; q++) cg[q] = 0.f;
    cg[0] = 1.f;                                                   // (0,0,0)
    for (int j = 0; j < 3; j++) cg[1 + j * 3 + j] = 1.f / s3;      // (0,1,1)
    for (int j = 0; j < 5; j++) cg[10 + j * 5 + j] = 1.f / s5;     // (0,2,2)
    for (int i = 0; i < 3; i++) cg[35 + i * 3 + i] = 1.f / s3;     // (1,0,1)
    for (int i = 0; i < 3; i++) cg[44 + i * 3 + i] = 1.f / s3;     // (1,1,0)
    for (int i = 0; i < 3; i++) for (int j = 0; j < 3; j++) for (int k = 0; k < 5; k++)
        cg[53 + (i * 3 + j) * 5 + k] = B[k][i][j] / s5;            // (1,1,2)
    for (int i = 0; i < 3; i++) for (int j = 0; j < 5; j++) for (int k = 0; k < 3; k++)
        cg[98 + (i * 5 + j) * 3 + k] = B[j][i][k] / s5;            // (1,2,1)
    for (int i = 0; i < 5; i++) cg[143 + i * 5 + i] = 1.f / s5;    // (2,0,2)
    for (int i = 0; i < 5; i++) for (int j = 0; j < 3; j++) for (int k = 0; k < 3; k++)
        cg[168 + (i * 3 + j) * 3 + k] = B[i][j][k] / s5;           // (2,1,1)
    for (int i = 0; i < 5; i++) cg[213 + i * 5 + i] = 1.f / s5;    // (2,2,0)
    float T[5][5][5]; float nrm = 0.f;
    for (int a = 0; a < 5; a++) for (int b = 0; b < 5; b++) for (int c = 0; c < 5; c++) {
        float s = 0.f;
        for (int i = 0; i < 3; i++) for (int j = 0; j < 3; j++) for (int k = 0; k < 3; k++)
            s += B[a][i][j] * B[b][j][k] * B[c][k][i];
        T[a][b][c] = s; nrm += s * s;
    }
    nrm = sqrtf(nrm);
    for (int a = 0; a < 5; a++) for (int b = 0; b < 5; b++) for (int c = 0; c < 5; c++)
        cg[238 + (a * 5 + b) * 5 + c] = T[a][b][c] / nrm;          // (2,2,2)
}

// ---------------------------------------------------------------------------
// h = node_features @ emb_w / SQM
// ---------------------------------------------------------------------------
__global__ void k_node_embed(const float* __restrict__ nf, const float* __restrict__ emb,
                             float* __restrict__ h) {
    int idx = blockIdx.x * blockDim.x + threadIdx.x;
    if (idx >= NNODES * MULC) return;
    int z = idx / MULC, w = idx % MULC;
    float s = 0.f;
    for (int u = 0; u < MULC; u++) s += nf[z * MULC + u] * emb[u * MULC + w];
    h[idx] = s / SQM;
}

// ---------------------------------------------------------------------------
// Radial MLP: rin(24) -> silu 64 -> silu 64 -> 64.  One 64-thread block/edge.
// ---------------------------------------------------------------------------
__global__ void k_radial(const float* __restrict__ remb, const float* __restrict__ attr,
                         const float* __restrict__ w1, const float* __restrict__ b1,
                         const float* __restrict__ w2, const float* __restrict__ b2,
                         const float* __restrict__ w3, const float* __restrict__ b3,
                         float* __restrict__ rf) {
    __shared__ float rin[24], a1[RHIDC], a2[RHIDC];
    int e = blockIdx.x, t = threadIdx.x;
    if (t < 8)       rin[t] = remb[e * 8 + t];
    else if (t < 24) rin[t] = attr[e * 16 + (t - 8)];
    __syncthreads();
    float s = b1[t];
    for (int q = 0; q < 24; q++) s += rin[q] * w1[q * RHIDC + t];
    a1[t] = s / (1.f + expf(-s));
    __syncthreads();
    s = b2[t];
    for (int q = 0; q < RHIDC; q++) s += a1[q] * w2[q * RHIDC + t];
    a2[t] = s / (1.f + expf(-s));
    __syncthreads();
    s = b3[t];
    for (int q = 0; q < RHIDC; q++) s += a2[q] * w3[q * RHIDC + t];
    rf[e * RHIDC + t] = s;
}

// ---------------------------------------------------------------------------
// Fused edge kernel (WMMA fp32): per 16-edge tile
//   tpw = rf @ a_w (+a_b)   [16 x 3072 GEMM, never materialized in memory]
//   mixed[e,l,w] = sum_u h_src[e,u]*tpw[e,l,u,w] / SQM   (epilogue-fused)
//   msg = mixed * sh / sqrt(d)  -> atomicAdd into a_sum[dst]; count into cnt.
// ---------------------------------------------------------------------------
__global__ __launch_bounds__(128)
void k_edge_msg(const float* __restrict__ rf, const float* __restrict__ h,
                const int* __restrict__ eidx, const float* __restrict__ aw,
                const float* __restrict__ ab, const float* __restrict__ sh,
                float* __restrict__ a_sum, float* __restrict__ cnt) {
    __shared__ float rfS[16][RHIDC];
    __shared__ float hS[16][MULC];
    __shared__ float mixedS[16][96];   // [edge][l*32+w]
    __shared__ int srcS[16], dstS[16];
    const int tid = threadIdx.x;
    const int e0 = blockIdx.x * 16;
    if (tid < 16) { srcS[tid] = eidx[e0 + tid]; dstS[tid] = eidx[NEDGES + e0 + tid]; }
    __syncthreads();
    for (int q = tid; q < 16 * RHIDC; q += 128) rfS[q / RHIDC][q % RHIDC] = rf[(e0 + q / RHIDC) * RHIDC + q % RHIDC];
    for (int q = tid; q < 16 * MULC; q += 128)  hS[q / MULC][q % MULC] = h[srcS[q / MULC] * MULC + q % MULC];
    for (int q = tid; q < 16 * 96; q += 128)    mixedS[q / 96][q % 96] = 0.f;
    __syncthreads();

    const int lane = tid & 31, wave = tid >> 5;
    const int half = (lane >> 4) & 1, ln = lane & 15;
    // 192 N-tiles of 16 cols over the 3072-wide (l,u,w) axis; w innermost so
    // each tile has fixed (l,u) and a 16-wide w range.
    for (int tile = wave; tile < 192; tile += 4) {
        const int c0 = tile * 16;
        const int l = c0 >> 10, u = (c0 & 1023) >> 5, w0 = c0 & 31;
        v8f acc = {0.f, 0.f, 0.f, 0.f, 0.f, 0.f, 0.f, 0.f};
        for (int ks = 0; ks < 16; ks++) {
            const int k0 = ks * 4;
            v2f a, b;
            a.x = rfS[ln][k0 + 2 * half];
            a.y = rfS[ln][k0 + 2 * half + 1];
            b.x = aw[(k0 + 2 * half) * 3072 + c0 + ln];
            b.y = aw[(k0 + 2 * half + 1) * 3072 + c0 + ln];
            acc = __builtin_amdgcn_wmma_f32_16x16x4_f32(false, a, false, b, (short)0, acc, false, false);
        }
#pragma unroll
        for (int jv = 0; jv < 8; jv++) {
            const int m = jv + 8 * half;                   // edge row in D
            atomicAdd(&mixedS[m][l * 32 + w0 + ln], acc[jv] * hS[m][u]);
        }
    }
    __syncthreads();
    // finalize: add bias path, scale, multiply by sh, scatter to a_sum[dst]
    for (int q = tid; q < 16 * 96; q += 128) {
        const int el = q / 96, c = q % 96, l = c >> 5, w = c & 31;
        float bias = 0.f;
        for (int u = 0; u < MULC; u++) bias += hS[el][u] * ab[l * 1024 + u * 32 + w];
        const int d = c_dims[l];
        const float mval = (mixedS[el][c] + bias) / SQM * c_dinv[l];
        const int dst = dstS[el];
        for (int i = 0; i < d; i++) {
            const float v = mval * sh[(e0 + el) * 9 + c_shoff[l] + i];
            atomicAdd(&a_sum[dst * HIDC + c_off[l] + w * d + i], v);
        }
    }
    if (tid < 16) atomicAdd(&cnt[dstS[tid]], 1.0f);
}

// a = a_sum / max(cnt,1)    (in place)
__global__ void k_pool_div(float* __restrict__ a, const float* __restrict__ cnt) {
    int idx = blockIdx.x * blockDim.x + threadIdx.x;
    if (idx >= NNODES * HIDC) return;
    a[idx] /= fmaxf(cnt[idx / HIDC], 1.0f);
}

// lin_irreps: out[z,w,i] = sum_u in[z,u,i] * W[l][u][w] / SQM
__global__ void k_lin(const float* __restrict__ in, const float* __restrict__ W,
                      float* __restrict__ out) {
    int idx = blockIdx.x * blockDim.x + threadIdx.x;
    if (idx >= NNODES * HIDC) return;
    const int z = idx / HIDC, c = idx % HIDC;
    int l, w, i, base;
    if (c < 32)       { l = 0; base = 0;   w = c;            i = 0; }
    else if (c < 128) { l = 1; base = 32;  w = (c - 32) / 3; i = (c - 32) % 3; }
    else              { l = 2; base = 128; w = (c - 128) / 5; i = (c - 128) % 5; }
    const int d = c_dims[l];
    float s = 0.f;
    for (int u = 0; u < MULC; u++) s += in[z * HIDC + base + u * d + i] * W[l * 1024 + u * 32 + w];
    out[idx] = s / SQM;
}

// ---------------------------------------------------------------------------
// uvw tensor-product path (WMMA fp32), templated so accumulators stay in VGPRs:
//   out[z,w,k] += scale * sum_{u,v} T[z,(u,v),k] * W[(u,v),w]
//   T[z,(u,v),k] = sum_{i,j} x[z,u,i]*y[z,v,j]*CG[i,j,k]
// 16-node tile, K=1024 in 16 blocks of 64; 2 waves own the two 16-wide w halves.
// ---------------------------------------------------------------------------
template <int DI, int DJ, int DK>
__global__ __launch_bounds__(64)
void k_tp(const float* __restrict__ X, int xoff, const float* __restrict__ Y, int yoff,
          const float* __restrict__ Wp, float* __restrict__ OUT, int koff,
          const float* __restrict__ cgp, float scale) {
    __shared__ float Xs[16 * 32 * DI];
    __shared__ float Ys[16 * 32 * DJ];
    __shared__ float Ts[DK][64 * 16];     // [k][kk*16 + z]
    __shared__ float Ws[64 * 32];         // [kk*32 + w]
    __shared__ float CGs[DI * DJ * DK];
    const int tid = threadIdx.x;
    const int z0 = blockIdx.x * 16;
    for (int q = tid; q < DI * DJ * DK; q += 64) CGs[q] = cgp[q];
    for (int q = tid; q < 16 * 32 * DI; q += 64) {
        const int zl = q / (32 * DI);
        Xs[q] = X[(z0 + zl) * HIDC + xoff + (q - zl * 32 * DI)];
    }
    for (int q = tid; q < 16 * 32 * DJ; q += 64) {
        const int zl = q / (32 * DJ);
        Ys[q] = Y[(z0 + zl) * HIDC + yoff + (q - zl * 32 * DJ)];
    }
    const int lane = tid & 31, wave = tid >> 5;
    const int half = (lane >> 4) & 1, ln = lane & 15;
    v8f acc[DK];
#pragma unroll
    for (int kc = 0; kc < DK; kc++) acc[kc] = (v8f){0.f, 0.f, 0.f, 0.f, 0.f, 0.f, 0.f, 0.f};

    for (int kb = 0; kb < 16; kb++) {
        __syncthreads();
        // build T block (64 uv values x 16 nodes x DK)
        for (int q = tid; q < 64 * 16; q += 64) {
            const int kk = q >> 4, zl = q & 15;
            const int u = kb * 2 + (kk >> 5), v = kk & 31;
            const float* xr = &Xs[zl * 32 * DI + u * DI];
            const float* yr = &Ys[zl * 32 * DJ + v * DJ];
            float tv[DK];
#pragma unroll
            for (int kc = 0; kc < DK; kc++) tv[kc] = 0.f;
#pragma unroll
            for (int ii = 0; ii < DI; ii++) {
                const float xv = xr[ii];
#pragma unroll
                for (int jj = 0; jj < DJ; jj++) {
                    const float xy = xv * yr[jj];
#pragma unroll
                    for (int kc = 0; kc < DK; kc++) tv[kc] += xy * CGs[(ii * DJ + jj) * DK + kc];
                }
            }
#pragma unroll
            for (int kc = 0; kc < DK; kc++) Ts[kc][kk * 16 + zl] = tv[kc];
        }
        // stage W block
        for (int q = tid; q < 64 * 32; q += 64) {
            const int kk = q >> 5, w = q & 31;
            Ws[q] = Wp[(kb * 2 + (kk >> 5)) * 1024 + (kk & 31) * 32 + w];
        }
        __syncthreads();
        // WMMA over the 64-wide K block
        for (int ks = 0; ks < 16; ks++) {
            const int k0 = ks * 4;
            v2f b;
            b.x = Ws[(k0 + 2 * half) * 32 + wave * 16 + ln];
            b.y = Ws[(k0 + 2 * half + 1) * 32 + wave * 16 + ln];
#pragma unroll
            for (int kc = 0; kc < DK; kc++) {
                v2f a;
                a.x = Ts[kc][(k0 + 2 * half) * 16 + ln];
                a.y = Ts[kc][(k0 + 2 * half + 1) * 16 + ln];
                acc[kc] = __builtin_amdgcn_wmma_f32_16x16x4_f32(false, a, false, b, (short)0, acc[kc], false, false);
            }
        }
    }
    // epilogue: D element (m,n): m = jv + 8*half, w = wave*16 + ln
#pragma unroll
    for (int kc = 0; kc < DK; kc++) {
#pragma unroll
        for (int jv = 0; jv < 8; jv++) {
            const int m = jv + 8 * half;
            const int w = wave * 16 + ln;
            atomicAdd(&OUT[(z0 + m) * HIDC + koff + w * DK + kc], acc[kc][jv] * scale);
        }
    }
}

// O2_UVU paths: out[z,u,k] += sum_{v,i,j} x[z,u,i]*x[z,v,j]*CG[i,j,k]*W[u,v] / SQM
__global__ void k_o2_uvu(const float* __restrict__ x, const float* __restrict__ W,
                         const float* __restrict__ cg, float* __restrict__ out) {
    int idx = blockIdx.x * blockDim.x + threadIdx.x;
    if (idx >= NNODES * MULC) return;
    const int z = idx / MULC, u = idx % MULC;
    float accum[9];
    for (int q = 0; q < 9; q++) accum[q] = 0.f;
    for (int p = 0; p < 5; p++) {
        const int li = c_uvu_i[p], lj = c_uvu_j[p], lk = c_uvu_k[p];
        const int di = c_dims[li], dj = c_dims[lj], dk = c_dims[lk];
        float R[5];
        for (int jj = 0; jj < dj; jj++) R[jj] = 0.f;
        for (int v = 0; v < MULC; v++) {
            const float wv = W[p * 1024 + u * 32 + v];
            const float* yr = &x[z * HIDC + c_off[lj] + v * dj];
            for (int jj = 0; jj < dj; jj++) R[jj] += wv * yr[jj];
        }
        const float* xr = &x[z * HIDC + c_off[li] + u * di];
        const float* cgp = &cg[c_uvu_cg[p]];
        const int ab = (lk == 0) ? 0 : ((lk == 1) ? 1 : 4);
        for (int ii = 0; ii < di; ii++)
            for (int jj = 0; jj < dj; jj++) {
                const float xy = xr[ii] * R[jj];
                for (int kc = 0; kc < dk; kc++)
                    accum[ab + kc] += xy * cgp[(ii * dj + jj) * dk + kc];
            }
    }
    out[z * HIDC + u] += accum[0] / SQM;
    for (int kc = 0; kc < 3; kc++) out[z * HIDC + 32 + u * 3 + kc] += accum[1 + kc] / SQM;
    for (int kc = 0; kc < 5; kc++) out[z * HIDC + 128 + u * 5 + kc] += accum[4 + kc] / SQM;
}

// fused mix+comb lin_irreps: out += comb_l( mix_l(in) ), block = one (z, l, i)
__global__ void k_mixcomb(const float* __restrict__ in, const float* __restrict__ mixW,
                          const float* __restrict__ combW, float* __restrict__ out) {
    __shared__ float t[MULC];
    const int z = blockIdx.x;
    const int i9 = blockIdx.y;
    const int l = (i9 == 0) ? 0 : ((i9 <= 3) ? 1 : 2);
    const int i = i9 - ((l == 0) ? 0 : ((l == 1) ? 1 : 4));
    const int d = c_dims[l], base = c_off[l];
    const int lane = threadIdx.x;
    float s = 0.f;
    for (int u = 0; u < MULC; u++) s += in[z * HIDC + base + u * d + i] * mixW[l * 1024 + u * 32 + lane];
    t[lane] = s / SQM;
    __syncthreads();
    float o = 0.f;
    for (int u = 0; u < MULC; u++) o += t[u] * combW[l * 1024 + u * 32 + lane];
    out[z * HIDC + base + lane * d + i] += o / SQM;
}

// self term: out[z, u (l0 slot)] += (nf @ self_w / SQM)
__global__ void k_self(const float* __restrict__ nf, const float* __restrict__ sw,
                       float* __restrict__ out) {
    int idx = blockIdx.x * blockDim.x + threadIdx.x;
    if (idx >= NNODES * MULC) return;
    const int z = idx / MULC, u = idx % MULC;
    float s = 0.f;
    for (int v = 0; v < MULC; v++) s += nf[z * MULC + v] * sw[v * MULC + u];
    out[z * HIDC + u] += s / SQM;
}

// ---------------------------------------------------------------------------
// Host side
// ---------------------------------------------------------------------------
static const int H_FULL_I[11]  = {0, 0, 0, 1, 1, 1, 1, 2, 2, 2, 2};
static const int H_FULL_J[11]  = {0, 1, 2, 0, 1, 1, 2, 0, 1, 2, 2};
static const int H_FULL_K[11]  = {0, 1, 2, 1, 0, 2, 1, 2, 1, 0, 2};
static const int H_FULL_CG[11] = {0, 1, 10, 35, 44, 53, 98, 143, 168, 213, 238};
static const int H_O2W[3]      = {1, 2, 6};   // (0,1,1),(0,2,2),(1,2,1)
static const int H_DIMS[3]     = {1, 3, 5};
static const int H_OFF[3]      = {0, 32, 128};

static void launch_tp(int di, int dj, int dk, const float* X, int xo, const float* Y, int yo,
                      const float* Wp, float* OUT, int ko, const float* cgp, float scale,
                      hipStream_t s) {
    dim3 g(NNODES / 16);
#define TP_CASE(a, b, c)                                                         \
    if (di == a && dj == b && dk == c) {                                         \
        k_tp<a, b, c><<<g, 64, 0, s>>>(X, xo, Y, yo, Wp, OUT, ko, cgp, scale);   \
        return;                                                                  \
    }
    TP_CASE(1, 1, 1) TP_CASE(1, 3, 3) TP_CASE(1, 5, 5) TP_CASE(3, 1, 3)
    TP_CASE(3, 3, 1) TP_CASE(3, 3, 5) TP_CASE(3, 5, 3) TP_CASE(5, 1, 5)
    TP_CASE(5, 3, 3) TP_CASE(5, 5, 1) TP_CASE(5, 5, 5)
#undef TP_CASE
}

extern "C" void kernel_launch(void* const* d_in, const int* in_sizes, int n_in,
                              void* d_out, int out_size, void* d_ws, size_t ws_size,
                              hipStream_t stream) {
    (void)in_sizes; (void)n_in; (void)ws_size;
    const float* nf    = (const float*)d_in[0];
    const int*   eidx  = (const int*)d_in[1];
    const float* sh    = (const float*)d_in[2];
    const float* remb  = (const float*)d_in[3];
    const float* attr  = (const float*)d_in[4];
    const float* rw1   = (const float*)d_in[5];
    const float* rb1   = (const float*)d_in[6];
    const float* rw2   = (const float*)d_in[7];
    const float* rb2   = (const float*)d_in[8];
    const float* rw3   = (const float*)d_in[9];
    const float* rb3   = (const float*)d_in[10];
    const float* a_w   = (const float*)d_in[11];
    const float* a_b   = (const float*)d_in[12];
    const float* emb_w = (const float*)d_in[13];
    const float* lin1  = (const float*)d_in[14];
    const float* o2uvw = (const float*)d_in[15];
    const float* o2uvu = (const float*)d_in[16];
    const float* o3a   = (const float*)d_in[17];
    const float* o3b   = (const float*)d_in[18];
    const float* mixw  = (const float*)d_in[19];
    const float* combw = (const float*)d_in[20];
    const float* selfw = (const float*)d_in[21];
    float* out = (float*)d_out;

    // workspace carve-up (all 256-aligned), total ~75.3 MB
    char* ws = (char*)d_ws;
    size_t o = 0;
    float* cg   = (float*)(ws + o); o += 4096;
    float* h    = (float*)(ws + o); o += (size_t)NNODES * MULC * 4;     // 1.28 MB
    float* rf   = (float*)(ws + o); o += (size_t)NEDGES * RHIDC * 4;    // 16.4 MB
    float* abuf = (float*)(ws + o); o += (size_t)NNODES * HIDC * 4;     // 11.5 MB
    float* cnt  = (float*)(ws + o); o += 40960;
    float* b1   = (float*)(ws + o); o += (size_t)NNODES * HIDC * 4;
    float* b2   = (float*)(ws + o); o += (size_t)NNODES * HIDC * 4;
    float* y1   = (float*)(ws + o); o += (size_t)NNODES * HIDC * 4;
    float* b3   = (float*)(ws + o); o += (size_t)NNODES * HIDC * 4;

    hipMemsetAsync(abuf, 0, (size_t)NNODES * HIDC * 4, stream);
    hipMemsetAsync(cnt,  0, 40960, stream);
    hipMemsetAsync(b2,   0, (size_t)NNODES * HIDC * 4, stream);
    hipMemsetAsync(y1,   0, (size_t)NNODES * HIDC * 4, stream);
    hipMemsetAsync(b3,   0, (size_t)NNODES * HIDC * 4, stream);
    hipMemsetAsync(out,  0, (size_t)out_size * 4, stream);

    k_init_cg<<<1, 1, 0, stream>>>(cg);
    k_node_embed<<<(NNODES * MULC + 255) / 256, 256, 0, stream>>>(nf, emb_w, h);
    k_radial<<<NEDGES, 64, 0, stream>>>(remb, attr, rw1, rb1, rw2, rb2, rw3, rb3, rf);
    k_edge_msg<<<NEDGES / 16, 128, 0, stream>>>(rf, h, eidx, a_w, a_b, sh, abuf, cnt);
    k_pool_div<<<(NNODES * HIDC + 255) / 256, 256, 0, stream>>>(abuf, cnt);

    // b1 = lin_irreps(a, lin_o1)
    k_lin<<<(NNODES * HIDC + 255) / 256, 256, 0, stream>>>(abuf, lin1, b1);

    // b2 = order2_tp(a): uvw paths via WMMA TP, uvu paths via VALU
    for (int p = 0; p < 3; p++) {
        const int pid = H_O2W[p];
        const int li = H_FULL_I[pid], lj = H_FULL_J[pid], lk = H_FULL_K[pid];
        launch_tp(H_DIMS[li], H_DIMS[lj], H_DIMS[lk], abuf, H_OFF[li], abuf, H_OFF[lj],
                  o2uvw + (size_t)p * 32768, b2, H_OFF[lk], cg + H_FULL_CG[pid],
                  1.0f / MULC, stream);
    }
    k_o2_uvu<<<(NNODES * MULC + 255) / 256, 256, 0, stream>>>(abuf, o2uvu, cg, b2);

    // b3 = pairwise(pairwise(a,a,o3a), a, o3b)
    for (int p = 0; p < 11; p++) {
        const int li = H_FULL_I[p], lj = H_FULL_J[p], lk = H_FULL_K[p];
        launch_tp(H_DIMS[li], H_DIMS[lj], H_DIMS[lk], abuf, H_OFF[li], abuf, H_OFF[lj],
                  o3a + (size_t)p * 32768, y1, H_OFF[lk], cg + H_FULL_CG[p],
                  1.0f / MULC, stream);
    }
    for (int p = 0; p < 11; p++) {
        const int li = H_FULL_I[p], lj = H_FULL_J[p], lk = H_FULL_K[p];
        launch_tp(H_DIMS[li], H_DIMS[lj], H_DIMS[lk], y1, H_OFF[li], abuf, H_OFF[lj],
                  o3b + (size_t)p * 32768, b3, H_OFF[lk], cg + H_FULL_CG[p],
                  1.0f / MULC, stream);
    }

    // msg = sum_order comb(mix(b_order)); + self term
    dim3 mcg(NNODES, 9);
    k_mixcomb<<<mcg, 32, 0, stream>>>(b1, mixw + 0 * 3072, combw + 0 * 3072, out);
    k_mixcomb<<<mcg, 32, 0, stream>>>(b2, mixw + 1 * 3072, combw + 1 * 3072, out);
    k_mixcomb<<<mcg, 32, 0, stream>>>(b3, mixw + 2 * 3072, combw + 2 * 3072, out);
    k_self<<<(NNODES * MULC + 255) / 256, 256, 0, stream>>>(nf, selfw, out);
}